// SelectiveDropout_85779086835903
// MI455X (gfx1250) — compile-verified
//
#include <hip/hip_runtime.h>
#include <hip/hip_bf16.h>

#define DCOLS 4096          // feature dimension D (columns)
#define DCOLS4 (DCOLS / 4)  // float4 columns

typedef float v4f __attribute__((ext_vector_type(4)));

// ---------------------------------------------------------------------------
// Threefry-2x32, exactly as JAX implements it (Random123 variant, 20 rounds).
// key = jax.random.key(42)  ->  (k0, k1) = (0, 42)
// ---------------------------------------------------------------------------
__device__ __forceinline__ unsigned rotl32(unsigned x, int d) {
    return (x << d) | (x >> (32 - d));
}

__device__ __forceinline__ void threefry2x32(unsigned k0, unsigned k1,
                                             unsigned& x0, unsigned& x1) {
    const unsigned ks0 = k0, ks1 = k1, ks2 = k0 ^ k1 ^ 0x1BD11BDAu;
    const int rA[4] = {13, 15, 26, 6};
    const int rB[4] = {17, 29, 16, 24};
    x0 += ks0; x1 += ks1;
#pragma unroll
    for (int i = 0; i < 4; i++) { x0 += x1; x1 = rotl32(x1, rA[i]); x1 ^= x0; }
    x0 += ks1; x1 += ks2 + 1u;
#pragma unroll
    for (int i = 0; i < 4; i++) { x0 += x1; x1 = rotl32(x1, rB[i]); x1 ^= x0; }
    x0 += ks2; x1 += ks0 + 2u;
#pragma unroll
    for (int i = 0; i < 4; i++) { x0 += x1; x1 = rotl32(x1, rA[i]); x1 ^= x0; }
    x0 += ks0; x1 += ks1 + 3u;
#pragma unroll
    for (int i = 0; i < 4; i++) { x0 += x1; x1 = rotl32(x1, rB[i]); x1 ^= x0; }
    x0 += ks1; x1 += ks2 + 4u;
#pragma unroll
    for (int i = 0; i < 4; i++) { x0 += x1; x1 = rotl32(x1, rA[i]); x1 ^= x0; }
    x0 += ks2; x1 += ks0 + 5u;
}

// ---------------------------------------------------------------------------
// Prep (single block, 256 threads = 8 waves): build scale[DCOLS] in one
// dispatch. winner[] lives in LDS; scatter uses ds_max atomics; phases are
// separated by workgroup barriers.
//   phase 1: winner[d] = -1
//   phase 2: per selected neuron i, draw u_i (JAX Threefry semantics),
//            keep-bit, last-write-wins via atomicMax(winner[col], 2*i+keep).
//            JAX even-size counter split: i<half -> (x0=i, x1=i+half, out0);
//            else (x0=i-half, x1=i, out1).
//   phase 3: scale[d] = -1 ? 1.0 : (keep ? 1/(1-p) : 0)
// ---------------------------------------------------------------------------
__global__ __launch_bounds__(256) void sd_prep(const int* __restrict__ idx,
                                               float* __restrict__ scale,
                                               int n_sel, float dropout_rate,
                                               float inv_keep) {
    __shared__ int winner[DCOLS];  // 16 KB

    for (int d = threadIdx.x; d < DCOLS; d += 256) winner[d] = -1;
    __syncthreads();

    const int half = n_sel >> 1;
    for (int i = threadIdx.x; i < n_sel; i += 256) {
        unsigned x0, x1;
        if (i < half) { x0 = (unsigned)i;          x1 = (unsigned)(i + half); }
        else          { x0 = (unsigned)(i - half); x1 = (unsigned)i; }
        threefry2x32(0u, 42u, x0, x1);
        unsigned bits = (i < half) ? x0 : x1;
        // JAX uniform: bitcast((bits>>9) | 0x3f800000) - 1.0f  in [0,1)
        float u = __uint_as_float((bits >> 9) | 0x3F800000u) - 1.0f;
        int keep = (u < dropout_rate) ? 0 : 1;
        int col = idx[i];
        if (col >= 0 && col < DCOLS) {
            atomicMax(&winner[col], i * 2 + keep);  // LDS ds_max
        }
    }
    __syncthreads();

    for (int d = threadIdx.x; d < DCOLS; d += 256) {
        int w = winner[d];
        scale[d] = (w < 0) ? 1.0f : ((w & 1) ? inv_keep : 0.0f);
    }
}

// ---------------------------------------------------------------------------
// Main: out = in * scale[col], bandwidth-bound streaming kernel.
//  - stage the 16 KB scale vector into LDS with the CDNA5 async global->LDS
//    path (global_load_async_to_lds_b128, ASYNCcnt, s_wait_asynccnt)
//  - grid stride (256 blk * 256 thr = 65536 float4) is a multiple of the 1024
//    float4-columns, so each thread's scale float4 is loop-invariant: one LDS
//    read, then pure non-temporal b128 load/mul/store streaming (input read
//    once, output never re-read -> keep the 128 MB stream out of L2).
// ---------------------------------------------------------------------------
__global__ __launch_bounds__(256) void sd_mul(const float* __restrict__ in,
                                              const float* __restrict__ scale,
                                              float* __restrict__ out, int n4) {
    __shared__ float s_scale[DCOLS];  // 16 KB

    const unsigned t = threadIdx.x;
    // Generic pointers to LDS carry the workgroup-relative byte offset in the
    // low 32 bits on AMDGCN.
    const unsigned lds_base = (unsigned)(unsigned long long)(&s_scale[0]);
#pragma unroll
    for (int k = 0; k < 4; k++) {
        unsigned byteoff = t * 16u + (unsigned)k * 4096u;  // 256 thr * 16 B * 4
        unsigned lds_addr = lds_base + byteoff;
        // GVS form: mem_addr = SADDR(64b) + VADDR(32b offset); VDST = LDS addr
        asm volatile("global_load_async_to_lds_b128 %0, %1, %2"
                     :
                     : "v"(lds_addr), "v"(byteoff), "s"(scale)
                     : "memory");
    }
    asm volatile("s_wait_asynccnt 0" ::: "memory");
    __syncthreads();

    const int stride = gridDim.x * blockDim.x;  // 65536 (multiple of DCOLS4)
    const int i0 = blockIdx.x * blockDim.x + t;
    const int c4 = i0 & (DCOLS4 - 1);  // loop-invariant float4 column
    const v4f s = *(const v4f*)&s_scale[c4 * 4];

    const v4f* __restrict__ in4 = (const v4f*)in;
    v4f* __restrict__ out4 = (v4f*)out;
    for (int i = i0; i < n4; i += stride) {
        v4f v = __builtin_nontemporal_load(&in4[i]);
        v *= s;
        __builtin_nontemporal_store(v, &out4[i]);
    }
}

// ---------------------------------------------------------------------------
extern "C" void kernel_launch(void* const* d_in, const int* in_sizes, int n_in,
                              void* d_out, int out_size, void* d_ws, size_t ws_size,
                              hipStream_t stream) {
    const float* inputs = (const float*)d_in[0];      // [B, D] fp32
    const int* neuron_indices = (const int*)d_in[1];  // [N_SEL] i32
    float* out = (float*)d_out;

    const int total = in_sizes[0];  // B * D
    const int n_sel = in_sizes[1];  // 2048
    const int n4 = total / 4;       // float4 elements

    float* scale = (float*)d_ws;    // scale[DCOLS] in workspace

    const float p = 0.1f;
    const float inv_keep = 1.0f / (1.0f - p);

    sd_prep<<<1, 256, 0, stream>>>(neuron_indices, scale, n_sel, p, inv_keep);
    sd_mul<<<256, 256, 0, stream>>>(inputs, scale, out, n4);
}